// GCN_with_feature_multilayers_60155311947857
// MI455X (gfx1250) — compile-verified
//
#include <hip/hip_runtime.h>
#include <hip/hip_bf16.h>

#define NN 100000
#define EE 1600000
#define HH 128
#define LDS_STRIDE 132   // 128 + 4 pad: rows hit distinct LDS bank groups

typedef __attribute__((ext_vector_type(2))) float v2f;
typedef __attribute__((ext_vector_type(8))) float v8f;

// ---------------------------------------------------------------- degrees ---
__global__ void deg_kernel(const int* __restrict__ src, const int* __restrict__ dst,
                           float* __restrict__ deg_out, float* __restrict__ deg_in) {
    int e = blockIdx.x * blockDim.x + threadIdx.x;
    if (e < EE) {
        atomicAdd(&deg_out[src[e]], 1.0f);
        atomicAdd(&deg_in[dst[e]], 1.0f);
    }
}

__global__ void norm_kernel(float* __restrict__ ns, float* __restrict__ nd) {
    int i = blockIdx.x * blockDim.x + threadIdx.x;
    if (i < NN) {
        ns[i] = rsqrtf(fmaxf(ns[i], 1.0f));
        nd[i] = rsqrtf(fmaxf(nd[i], 1.0f));
    }
}

// ------------------------------------------------------- gather/scatter-add --
// One wave per edge: 32 lanes x float4 = 128 feats. agg is L2-resident, so the
// f32 atomics RMW in L2.
__global__ void scatter_edges(const float* __restrict__ feat,
                              const float* __restrict__ norm_src,
                              const int* __restrict__ src, const int* __restrict__ dst,
                              float* __restrict__ agg) {
    int e = blockIdx.x * (blockDim.x >> 5) + (threadIdx.x >> 5);
    if (e >= EE) return;
    int lane = threadIdx.x & 31;
    int s = src[e];
    int d = dst[e];
    float ns = norm_src[s];
    float4 v = ((const float4*)(feat + (size_t)s * HH))[lane];
    v.x *= ns; v.y *= ns; v.z *= ns; v.w *= ns;
    float* ad = agg + (size_t)d * HH + lane * 4;
    atomicAdd(ad + 0, v.x);
    atomicAdd(ad + 1, v.y);
    atomicAdd(ad + 2, v.z);
    atomicAdd(ad + 3, v.w);
}

// ------------------------------------------------------- fused GEMM + epilog --
// out[m,n] = sum_k (agg[m,k]*norm_dst[m]) * W[k,n] + bias[n] (+ res*ori[m,n])
// Block = 8 waves; block owns 16 rows (A tile staged in LDS, pre-scaled by
// norm_dst), each wave owns one 16x16 output tile, K-loop = 32 chained
// v_wmma_f32_16x16x4_f32.  Grid is exact -> EXEC all-ones for every WMMA.
__global__ void gcn_gemm(const float* __restrict__ agg,
                         const float* __restrict__ norm_dst,
                         const float* __restrict__ W,
                         const float* __restrict__ bias,
                         const float* __restrict__ ori_in,   // may be null
                         float res,
                         float* __restrict__ out,
                         float* __restrict__ ori_out) {      // may be null
    __shared__ float lds_a[16 * LDS_STRIDE];

    const int m0   = blockIdx.x * 16;
    const int tid  = threadIdx.x;       // 0..255
    const int wave = tid >> 5;          // 0..7 -> 16-col tile
    const int lane = tid & 31;

    // Cooperative load of the 16x128 A tile, scaled by norm_dst.
    {
        int base = tid * 8;             // 8 floats per thread
        int row  = base >> 7;           // /128
        int col  = base & 127;
        float nd = norm_dst[m0 + row];
        const float4* s4 = (const float4*)(agg + (size_t)(m0 + row) * HH + col);
        float4 a0 = s4[0];
        float4 a1 = s4[1];
        a0.x *= nd; a0.y *= nd; a0.z *= nd; a0.w *= nd;
        a1.x *= nd; a1.y *= nd; a1.z *= nd; a1.w *= nd;
        float4* t4 = (float4*)(lds_a + row * LDS_STRIDE + col);
        t4[0] = a0;
        t4[1] = a1;
    }
    __syncthreads();

    const int n0   = wave * 16;
    const int half = lane >> 4;         // 0: K+{0,1}, 1: K+{2,3}
    const int l15  = lane & 15;

    // A fragment source: row l15, K offset (half*2)
    const float* arow = lds_a + l15 * LDS_STRIDE + half * 2;
    // B fragment source: W[k][n], n = n0+l15, K offset (half*2)
    const float* bcol = W + (size_t)(half * 2) * HH + n0 + l15;

    v8f c = {};
#pragma unroll
    for (int kk = 0; kk < 32; ++kk) {
        const int k0 = kk * 4;
        v2f a, b;
        a.x = arow[k0];
        a.y = arow[k0 + 1];
        b.x = bcol[(size_t)k0 * HH];
        b.y = bcol[(size_t)k0 * HH + HH];
        c = __builtin_amdgcn_wmma_f32_16x16x4_f32(
                /*neg_a=*/false, a, /*neg_b=*/false, b,
                /*c_mod=*/(short)0, c, /*reuse_a=*/false, /*reuse_b=*/false);
    }

    // Epilogue: C/D layout -> VGPR r: lanes 0-15 M=r, lanes 16-31 M=r+8; N=l15.
    const int nn = n0 + l15;
    const float bv = bias[nn];
#pragma unroll
    for (int r = 0; r < 8; ++r) {
        int m = m0 + r + half * 8;
        size_t idx = (size_t)m * HH + nn;
        float v = c[r] + bv;
        if (ori_in)  v += res * ori_in[idx];
        out[idx] = v;
        if (ori_out) ori_out[idx] = v;
    }
}

// ------------------------------------------------------- layernorm + relu ----
// One wave per node (32 lanes x float4 = 128 feats), in-place.
__global__ void ln_relu(float* __restrict__ h,
                        const float* __restrict__ g,
                        const float* __restrict__ bta) {
    int n = blockIdx.x * (blockDim.x >> 5) + (threadIdx.x >> 5);
    if (n >= NN) return;
    int lane = threadIdx.x & 31;
    float4* row = (float4*)(h + (size_t)n * HH);
    float4 v = row[lane];

    float s = v.x + v.y + v.z + v.w;
#pragma unroll
    for (int off = 16; off > 0; off >>= 1) s += __shfl_xor(s, off, 32);
    float mu = s * (1.0f / 128.0f);

    float dx = v.x - mu, dy = v.y - mu, dz = v.z - mu, dw = v.w - mu;
    float q = dx * dx + dy * dy + dz * dz + dw * dw;
#pragma unroll
    for (int off = 16; off > 0; off >>= 1) q += __shfl_xor(q, off, 32);
    float r = rsqrtf(q * (1.0f / 128.0f) + 1e-5f);

    int c = lane * 4;
    float4 o;
    o.x = fmaxf(dx * r * g[c + 0] + bta[c + 0], 0.0f);
    o.y = fmaxf(dy * r * g[c + 1] + bta[c + 1], 0.0f);
    o.z = fmaxf(dz * r * g[c + 2] + bta[c + 2], 0.0f);
    o.w = fmaxf(dw * r * g[c + 3] + bta[c + 3], 0.0f);
    row[lane] = o;
}

// ----------------------------------------------------------------- driver ----
extern "C" void kernel_launch(void* const* d_in, const int* in_sizes, int n_in,
                              void* d_out, int out_size, void* d_ws, size_t ws_size,
                              hipStream_t stream) {
    (void)in_sizes; (void)n_in; (void)out_size; (void)ws_size;

    const float* x    = (const float*)d_in[0];   // [N,128]
    const float* W0   = (const float*)d_in[1];   // [128,128]
    const float* Ws   = (const float*)d_in[2];   // [2,128,128]
    const float* b    = (const float*)d_in[3];   // [3,128]
    const float* ln_g = (const float*)d_in[4];   // [3,128]
    const float* ln_b = (const float*)d_in[5];   // [3,128]
    const int*   src  = (const int*)d_in[6];     // [E]
    const int*   dst  = (const int*)d_in[7];     // [E]
    float* out = (float*)d_out;                  // [N,128]

    float* ws       = (float*)d_ws;
    float* norm_src = ws;                        // N
    float* norm_dst = ws + NN;                   // N
    float* agg      = ws + 2 * (size_t)NN;       // N*H
    float* ori      = agg + (size_t)NN * HH;     // N*H
    float* hbuf     = ori + (size_t)NN * HH;     // N*H

    const size_t featBytes = (size_t)NN * HH * sizeof(float);

    // Degrees -> norms
    hipMemsetAsync(ws, 0, 2 * (size_t)NN * sizeof(float), stream);
    deg_kernel<<<EE / 256, 256, 0, stream>>>(src, dst, norm_src, norm_dst);
    norm_kernel<<<(NN + 255) / 256, 256, 0, stream>>>(norm_src, norm_dst);

    // ---- Layer 0: h = GCN(x, W0) + b0 ; ori = h
    hipMemsetAsync(agg, 0, featBytes, stream);
    scatter_edges<<<EE / 8, 256, 0, stream>>>(x, norm_src, src, dst, agg);
    gcn_gemm<<<NN / 16, 256, 0, stream>>>(agg, norm_dst, W0, b, nullptr, 0.0f,
                                          hbuf, ori);

    // ---- Layer 1: h = GCN(relu(LN(h)), Ws[0]) + b1 + 0.5*ori
    ln_relu<<<NN / 8, 256, 0, stream>>>(hbuf, ln_g + HH, ln_b + HH);
    hipMemsetAsync(agg, 0, featBytes, stream);
    scatter_edges<<<EE / 8, 256, 0, stream>>>(hbuf, norm_src, src, dst, agg);
    gcn_gemm<<<NN / 16, 256, 0, stream>>>(agg, norm_dst, Ws, b + HH, ori, 0.5f,
                                          hbuf, nullptr);

    // ---- Layer 2: out = GCN(relu(LN(h)), Ws[1]) + b2 + 0.5*ori
    ln_relu<<<NN / 8, 256, 0, stream>>>(hbuf, ln_g + 2 * HH, ln_b + 2 * HH);
    hipMemsetAsync(agg, 0, featBytes, stream);
    scatter_edges<<<EE / 8, 256, 0, stream>>>(hbuf, norm_src, src, dst, agg);
    gcn_gemm<<<NN / 16, 256, 0, stream>>>(agg, norm_dst, Ws + (size_t)HH * HH,
                                          b + 2 * HH, ori, 0.5f, out, nullptr);
}